// FeaturesLinear_20040317403342
// MI455X (gfx1250) — compile-verified
//
#include <hip/hip_runtime.h>
#include <hip/hip_bf16.h>

typedef __attribute__((ext_vector_type(2))) float v2f;
typedef __attribute__((ext_vector_type(8))) float v8f;

#define TOTAL   819200
#define BATCH   16384
#define DIM     16
#define CHUNK   64            // tokens per wave
#define WAVES_PER_BLOCK 8     // 256 threads, wave32

// out[b][d] = bias[d]  (re-initialized every launch; atomics accumulate on top)
__global__ void fl_init_kernel(float* __restrict__ out,
                               const float* __restrict__ bias, int n) {
    int i = blockIdx.x * blockDim.x + threadIdx.x;
    if (i < n) out[i] = bias[i & (DIM - 1)];
}

// One wave handles CHUNK consecutive tokens. Segment sum done on the matrix
// pipe: C(16 local segs x 16 dims) += A(16x4 one-hot*rating) x B(4x16 rows).
__global__ void __launch_bounds__(256)
fl_wmma_kernel(const int*   __restrict__ feature_ids,
               const float* __restrict__ feature_ratings,
               const int*   __restrict__ segment_ids,
               const float* __restrict__ weight,
               float*       __restrict__ out) {
    const int lane = threadIdx.x & 31;
    // readfirstlane makes the chunk index *provably* wave-uniform so the
    // per-step metadata loads qualify for SMEM (s_load_b128) promotion.
    const int chunk = __builtin_amdgcn_readfirstlane(
        (int)(blockIdx.x * WAVES_PER_BLOCK + (threadIdx.x >> 5)));
    const int base  = chunk * CHUNK;

    const bool hi = (lane & 16) != 0;  // which 16-lane half (K-slot selector)
    const int  n  = lane & 15;         // A: M (local segment) / B,C: N (dim)

    // Wave-uniform segment span of this chunk (segment_ids sorted ascending).
    const int seg_lo = __builtin_amdgcn_readfirstlane(segment_ids[base]);
    const int seg_hi = __builtin_amdgcn_readfirstlane(segment_ids[base + CHUNK - 1]);
    const int span   = seg_hi - seg_lo;

    if (span < 16) {
        // -------- WMMA segment-sum path (common case) --------
        v8f c = {};
        #pragma unroll 2
        for (int step = 0; step < CHUNK / 4; ++step) {
            const int tbase = base + step * 4;  // 16B-aligned, scalar-uniform
            // Uniform 16B metadata loads -> constant cache (SMEM) candidates.
            const int4   seg4 = *(const int4*)  (segment_ids     + tbase);
            const float4 rat4 = *(const float4*)(feature_ratings + tbase);
            const int4   id4  = *(const int4*)  (feature_ids     + tbase);

            // This lane's two K slots: k = hi*2 + {0,1} (ISA f32 A/B layout).
            const int   sA = hi ? seg4.z : seg4.x;
            const int   sB = hi ? seg4.w : seg4.y;
            const float rA = hi ? rat4.z : rat4.x;
            const float rB = hi ? rat4.w : rat4.y;
            const int   iA = hi ? id4.z  : id4.x;
            const int   iB = hi ? id4.w  : id4.y;

            v2f a, b;
            // A[m][k] = rating iff token's local segment == m (rating folded in)
            a[0] = (sA - seg_lo == n) ? rA : 0.0f;
            a[1] = (sB - seg_lo == n) ? rB : 0.0f;
            // B[k][n] = gathered weight row element; 32-bit non-negative offset
            // (id < 2^20) so the gather uses SADDR + i32 VGPR addressing.
            b[0] = weight[((iA & 0xFFFFF) << 4) | n];
            b[1] = weight[((iB & 0xFFFFF) << 4) | n];

            c = __builtin_amdgcn_wmma_f32_16x16x4_f32(
                    /*neg_a=*/false, a, /*neg_b=*/false, b,
                    /*c_mod=*/(short)0, c, /*reuse_a=*/false, /*reuse_b=*/false);
        }
        // Merge local segment sums into global output (segments span chunks).
        #pragma unroll
        for (int r = 0; r < 8; ++r) {
            const int m = r + (hi ? 8 : 0);     // C row = local segment
            if (m <= span) {
                const int off = (((seg_lo + m) & 0x3FFF) << 4) | n;
                atomicAdd(&out[off], c[r]);
            }
        }
    } else {
        // -------- rare wide-span fallback: 2 tokens x 16 dims per iter --------
        const int tt = hi ? 1 : 0;
        for (int i = 0; i < CHUNK; i += 2) {
            const int t = base + i + tt;
            const int   seg = segment_ids[t];
            const float rat = feature_ratings[t];
            const int   id  = feature_ids[t];
            const int   woff = ((id & 0xFFFFF) << 4) | n;
            const int   ooff = ((seg & 0x3FFF) << 4) | n;
            atomicAdd(&out[ooff], weight[woff] * rat);
        }
    }
}

extern "C" void kernel_launch(void* const* d_in, const int* in_sizes, int n_in,
                              void* d_out, int out_size, void* d_ws, size_t ws_size,
                              hipStream_t stream) {
    const int*   feature_ids     = (const int*)  d_in[0];
    const float* feature_ratings = (const float*)d_in[1];
    const int*   segment_ids     = (const int*)  d_in[2];
    // d_in[3] = batch_size scalar (compile-time constant BATCH here)
    const float* weight          = (const float*)d_in[4];
    const float* bias            = (const float*)d_in[5];
    float* out = (float*)d_out;

    const int n_out = BATCH * DIM;                   // 262144
    fl_init_kernel<<<(n_out + 255) / 256, 256, 0, stream>>>(out, bias, n_out);

    const int n_chunks = TOTAL / CHUNK;              // 12800 waves
    const int n_blocks = n_chunks / WAVES_PER_BLOCK; // 1600 blocks
    fl_wmma_kernel<<<n_blocks, 256, 0, stream>>>(
        feature_ids, feature_ratings, segment_ids, weight, out);
}